// DRew_GIN_Layer_85031762526642
// MI455X (gfx1250) — compile-verified
//
#include <hip/hip_runtime.h>

typedef float v2f __attribute__((ext_vector_type(2)));
typedef float v8f __attribute__((ext_vector_type(8)));

#define CCH 128   // channels
#define NT_TILES 8
#define KT_STEPS 32

__device__ __forceinline__ v8f wmma_f32(v2f a, v2f b, v8f c) {
    // D = A(16x4 f32) x B(4x16 f32) + C(16x16 f32)
    return __builtin_amdgcn_wmma_f32_16x16x4_f32(
        /*neg_a=*/false, a, /*neg_b=*/false, b,
        /*c_mod=*/(short)0, c, /*reuse_a=*/false, /*reuse_b=*/false);
}

// Transpose the eight 128x128 weight matrices into ws so the GEMM's B-operand
// fragments become contiguous b64 loads. Layout in wsWT:
//   [0..3]  = W1^T for mlp m = 0(rel), 1(hh0), 2(hh1), 3(loop)
//   [4..7]  = W2^T for the same order
__global__ __launch_bounds__(256)
void drew_gin_transpose_w_kernel(const float* __restrict__ relW1,  const float* __restrict__ relW2,
                                 const float* __restrict__ hhW1,   const float* __restrict__ hhW2,
                                 const float* __restrict__ loopW1, const float* __restrict__ loopW2,
                                 float* __restrict__ wsWT)
{
    const int m = blockIdx.y;                  // 0..7
    const float* W;
    switch (m) {
        case 0: W = relW1;            break;
        case 1: W = hhW1;             break;
        case 2: W = hhW1 + CCH * CCH; break;
        case 3: W = loopW1;           break;
        case 4: W = relW2;            break;
        case 5: W = hhW2;             break;
        case 6: W = hhW2 + CCH * CCH; break;
        default: W = loopW2;          break;
    }
    const int idx = blockIdx.x * 256 + threadIdx.x;   // 0..16383, coalesced read
    const int k = idx >> 7;
    const int n = idx & 127;
    wsWT[(size_t)m * CCH * CCH + n * CCH + k] = W[idx];
}

// One wave computes a 16-row x 128-col tile of  out = relu(X@W1+b1)@W2+b2.
// 8 waves per block, block row-tile = 128 rows. blockIdx.y selects the MLP.
__global__ __launch_bounds__(256)
void drew_gin_mlp_kernel(const float* __restrict__ emb,     // [3, N, 128]
                         const float* __restrict__ wsWT,    // [8, 128, 128] transposed weights
                         const float* __restrict__ loopB1, const float* __restrict__ loopB2,
                         const float* __restrict__ relB1,  const float* __restrict__ relB2,
                         const float* __restrict__ hhB1,   const float* __restrict__ hhB2,
                         float* __restrict__ hopbuf,       // ws: [3, N, 128]
                         float* __restrict__ out,          // [N, 128] (self term)
                         int N)
{
    extern __shared__ float smem[];            // 8 waves * 16*128 floats = 64 KB
    const int lane = threadIdx.x & 31;
    const int wave = threadIdx.x >> 5;
    const int m    = blockIdx.y;
    const int r0   = blockIdx.x * 128 + wave * 16;
    if (r0 >= N) return;                       // wave-uniform: EXEC stays all-1s below

    const float *X, *B1, *B2;
    float* O;
    switch (m) {
        case 0:  // hop-1 relational MLP on x_t = emb[2]
            X = emb + 2 * (size_t)N * CCH;
            B1 = relB1; B2 = relB2;
            O = hopbuf; break;
        case 1:  // hop-2 MLP (hh idx 0) on emb[1]
            X = emb + 1 * (size_t)N * CCH;
            B1 = hhB1; B2 = hhB2;
            O = hopbuf + (size_t)N * CCH; break;
        case 2:  // hop-3 MLP (hh idx 1) on emb[0]
            X = emb;
            B1 = hhB1 + CCH; B2 = hhB2 + CCH;
            O = hopbuf + 2 * (size_t)N * CCH; break;
        default: // self-loop MLP on x_t, straight to d_out
            X = emb + 2 * (size_t)N * CCH;
            B1 = loopB1; B2 = loopB2;
            O = out; break;
    }
    const float* W1T = wsWT + (size_t)m * CCH * CCH;        // W1^T [n][k]
    const float* W2T = wsWT + (size_t)(4 + m) * CCH * CCH;  // W2^T [n][k]

    float* H = smem + wave * (16 * CCH);       // per-wave ReLU intermediate tile

    const int half = lane >> 4;                // fragment half: K offset +2
    const int l15  = lane & 15;
    int arow = r0 + l15;
    if (arow >= N) arow = N - 1;               // clamp so loads keep EXEC all-1s

    // ---------- GEMM1: H = relu(X@W1 + b1) ----------
    v2f a[KT_STEPS];
    const v2f* X2 = (const v2f*)(X + (size_t)arow * CCH);
    #pragma unroll
    for (int kt = 0; kt < KT_STEPS; ++kt)
        a[kt] = X2[kt * 2 + half];             // A[M=l15][k0..k0+1], k0 = 4*kt + 2*half

    #pragma unroll
    for (int nt = 0; nt < NT_TILES; ++nt) {
        v8f acc = {0.f,0.f,0.f,0.f,0.f,0.f,0.f,0.f};
        const int n = nt * 16 + l15;
        const v2f* W1Tn = (const v2f*)(W1T + (size_t)n * CCH);
        #pragma unroll
        for (int kt = 0; kt < KT_STEPS; ++kt) {
            v2f b = W1Tn[kt * 2 + half];       // B[k0..k0+1][n] as one b64 load
            acc = wmma_f32(a[kt], b, acc);
        }
        const float bias = B1[n];
        #pragma unroll
        for (int r = 0; r < 8; ++r) {          // D: VGPR r, M = r + 8*half, N = l15
            const int mrow = r + half * 8;
            float v = acc[r] + bias;
            H[mrow * CCH + n] = v > 0.f ? v : 0.f;
        }
    }

    // ---------- GEMM2: O = H@W2 + b2 (A fragments re-gathered from LDS) ----------
    #pragma unroll
    for (int kt = 0; kt < KT_STEPS; ++kt) {
        const int k0 = kt * 4 + half * 2;
        a[kt].x = H[l15 * CCH + k0];
        a[kt].y = H[l15 * CCH + k0 + 1];
    }
    #pragma unroll
    for (int nt = 0; nt < NT_TILES; ++nt) {
        v8f acc = {0.f,0.f,0.f,0.f,0.f,0.f,0.f,0.f};
        const int n = nt * 16 + l15;
        const v2f* W2Tn = (const v2f*)(W2T + (size_t)n * CCH);
        #pragma unroll
        for (int kt = 0; kt < KT_STEPS; ++kt) {
            v2f b = W2Tn[kt * 2 + half];
            acc = wmma_f32(a[kt], b, acc);
        }
        const float bias = B2[n];
        #pragma unroll
        for (int r = 0; r < 8; ++r) {
            const int grow = r0 + r + half * 8;
            if (grow < N)
                O[(size_t)grow * CCH + n] = acc[r] + bias;
        }
    }
}

// One wave per edge: out[row] += hop_coef[w-1] * hop[w-1][col]. Hop features
// (76.8 MB) and out (25.6 MB) are L2-resident (192 MB L2), so the random
// gather + atomics never touch HBM.
__global__ __launch_bounds__(256)
void drew_gin_scatter_kernel(const int* __restrict__ ei,       // [2, E]
                             const int* __restrict__ ew,       // [E]
                             const float* __restrict__ hop_coef,
                             const float* __restrict__ hopbuf, // [3, N, 128]
                             float* __restrict__ out, int N, int E)
{
    const int lane = threadIdx.x & 31;
    const int e = blockIdx.x * 8 + (threadIdx.x >> 5);
    if (e >= E) return;
    const int row = ei[e];
    const int col = ei[E + e];
    const int w   = ew[e];                     // 1..3
    const float coef = hop_coef[w - 1];
    const float4 v = ((const float4*)(hopbuf + ((size_t)(w - 1) * N + col) * CCH))[lane];
    float* dst = out + (size_t)row * CCH + lane * 4;
    unsafeAtomicAdd(dst + 0, coef * v.x);
    unsafeAtomicAdd(dst + 1, coef * v.y);
    unsafeAtomicAdd(dst + 2, coef * v.z);
    unsafeAtomicAdd(dst + 3, coef * v.w);
}

extern "C" void kernel_launch(void* const* d_in, const int* in_sizes, int n_in,
                              void* d_out, int out_size, void* d_ws, size_t ws_size,
                              hipStream_t stream) {
    const float* emb    = (const float*)d_in[1];
    const int*   ei     = (const int*)  d_in[2];
    const int*   ew     = (const int*)  d_in[3];
    const float* loopW1 = (const float*)d_in[4];
    const float* loopB1 = (const float*)d_in[5];
    const float* loopW2 = (const float*)d_in[6];
    const float* loopB2 = (const float*)d_in[7];
    const float* relW1  = (const float*)d_in[8];
    const float* relB1  = (const float*)d_in[9];
    const float* relW2  = (const float*)d_in[10];
    const float* relB2  = (const float*)d_in[11];
    const float* hhW1   = (const float*)d_in[12];
    const float* hhB1   = (const float*)d_in[13];
    const float* hhW2   = (const float*)d_in[14];
    const float* hhB2   = (const float*)d_in[15];
    const float* hcoef  = (const float*)d_in[16];

    const int N = in_sizes[1] / (3 * CCH);     // 50000
    const int E = in_sizes[3];                 // 600000

    float* hopbuf = (float*)d_ws;                          // 3*N*128 f32 = 76.8 MB
    float* wsWT   = hopbuf + 3 * (size_t)N * CCH;          // 8*128*128 f32 = 512 KB
    float* out    = (float*)d_out;

    // 1) transpose weights into ws (L2-resident thereafter)
    drew_gin_transpose_w_kernel<<<dim3(64, 8), 256, 0, stream>>>(
        relW1, relW2, hhW1, hhW2, loopW1, loopW2, wsWT);

    // 2) four fused MLPs via f32 WMMA; self term lands directly in d_out
    dim3 gemm_grid((N + 127) / 128, 4);
    drew_gin_mlp_kernel<<<gemm_grid, 256, 8 * 16 * CCH * sizeof(float), stream>>>(
        emb, wsWT, loopB1, loopB2, relB1, relB2, hhB1, hhB2,
        hopbuf, out, N);

    // 3) edge scatter with hardware f32 atomics (L2-resident)
    drew_gin_scatter_kernel<<<(E + 7) / 8, 256, 0, stream>>>(
        ei, ew, hcoef, hopbuf, out, N, E);
}